// GNN_88545045775118
// MI455X (gfx1250) — compile-verified
//
#include <hip/hip_runtime.h>
#include <hip/hip_bf16.h>

#define N_NODES 50000
#define N_EDGES 800000
#define D 128
#define EPS_LN 1e-5f
#define EPS_BN 1e-5f
#define SLOPE 0.01f

// LDS layout for the GEMM block (dword units). TDM pads +2 dwords every 128
// dwords -> row stride 130 (bank-conflict-free, keeps 8B alignment for b64).
#define ROWSTRIDE 130
#define SM_A 0                         // 16 rows  * 130 = 2080 dwords
#define SM_G (SM_A + 16 * ROWSTRIDE)   // 128 rows * 130 = 16640 dwords
#define SM_R (SM_G + 128 * ROWSTRIDE)
#define SM_TOTAL_DW (SM_R + 128 * ROWSTRIDE)     // 35360 dwords
#define SM_TOTAL_BYTES (SM_TOTAL_DW * 4)         // 141440 B

typedef float v2f  __attribute__((ext_vector_type(2)));
typedef float v8f  __attribute__((ext_vector_type(8)));
typedef unsigned int u32x4 __attribute__((ext_vector_type(4)));
typedef int   i32x4 __attribute__((ext_vector_type(4)));
typedef int   i32x8 __attribute__((ext_vector_type(8)));

__device__ __forceinline__ float wave_sum32(float s) {
#pragma unroll
    for (int off = 16; off > 0; off >>= 1) s += __shfl_xor(s, off, 32);
    return s;
}

// ---- TDM: issue a 2-D tile load (row-major fp32) into LDS with row padding.
// D# packing per CDNA5 ISA §8.3 (group 0) / §8.4 (group 1); groups 2-3 NULLed
// (zero) for a 2-D tensor.  data_size=4B, pad every 128 dwords by 2 dwords.
__device__ __forceinline__ void tdm_load_2d_f32(
    unsigned int lds_byte_addr, const float* gptr,
    unsigned int dim0, unsigned int dim1,        // tensor dims (elements)
    unsigned int tile0, unsigned int tile1,      // tile dims (elements)
    unsigned int stride0)                        // dim0 stride (elements)
{
    unsigned long long ga = (unsigned long long)(uintptr_t)gptr;
    u32x4 g0;
    g0.x = 1u;                                       // count=1, user mode
    g0.y = lds_byte_addr;                            // lds_addr
    g0.z = (unsigned int)(ga & 0xFFFFFFFFu);         // global_addr[31:0]
    g0.w = (unsigned int)((ga >> 32) & 0x01FFFFFFu)  // global_addr[56:32]
         | (2u << 30);                               // type=2 (image)

    i32x8 g1;
    g1[0] = (int)((2u << 16)        // data_size = 4 bytes
                | (1u << 20)        // pad_enable
                | (6u << 22)        // pad_interval: 128 dwords
                | (1u << 25));      // pad_amount: 2 dwords
    g1[1] = (int)((dim0 & 0xFFFFu) << 16);                    // tensor_dim0 lo
    g1[2] = (int)((dim0 >> 16) | ((dim1 & 0xFFFFu) << 16));   // dim0 hi | dim1 lo
    g1[3] = (int)((dim1 >> 16) | (tile0 << 16));              // dim1 hi | tile_dim0
    g1[4] = (int)(tile1 & 0xFFFFu);                           // tile_dim1 (tile_dim2=0)
    g1[5] = (int)stride0;                                     // tensor_dim0_stride lo32
    g1[6] = 0;                                                // stride0 hi | stride1 lo
    g1[7] = 0;                                                // stride1 hi

    i32x4 gz4 = {0, 0, 0, 0};
    i32x8 gz8 = {0, 0, 0, 0, 0, 0, 0, 0};
    __builtin_amdgcn_tensor_load_to_lds(g0, g1, gz4, gz4, gz8, 0);
}

// ---------------------------------------------------------------- init
__global__ void k_init(float* __restrict__ deg, float* __restrict__ bnsum,
                       float* __restrict__ bnsq) {
    int i = blockIdx.x * 256 + threadIdx.x;
    if (i < N_NODES) deg[i] = 1.0f;           // self-loop contributes 1 to degree
    if (i < D) { bnsum[i] = 0.0f; bnsq[i] = 0.0f; }
}

// ---------------------------------------------------------------- LayerNorm
// one wave32 per node row of 128 floats (float4 per lane)
__global__ void __launch_bounds__(256) k_layernorm(
    const float* __restrict__ x, const float* __restrict__ g,
    const float* __restrict__ b, float* __restrict__ xl) {
    const int node = blockIdx.x * 8 + (threadIdx.x >> 5);
    const int lane = threadIdx.x & 31;
    const float4 v = ((const float4*)(x + (size_t)node * D))[lane];
    const float mu = wave_sum32(v.x + v.y + v.z + v.w) * (1.0f / D);
    const float4 c = make_float4(v.x - mu, v.y - mu, v.z - mu, v.w - mu);
    const float var = wave_sum32(c.x * c.x + c.y * c.y + c.z * c.z + c.w * c.w) * (1.0f / D);
    const float inv = rsqrtf(var + EPS_LN);
    const float4 gg = ((const float4*)g)[lane];
    const float4 bb = ((const float4*)b)[lane];
    float4 o = make_float4(gg.x * c.x * inv + bb.x, gg.y * c.y * inv + bb.y,
                           gg.z * c.z * inv + bb.z, gg.w * c.w * inv + bb.w);
    ((float4*)(xl + (size_t)node * D))[lane] = o;
}

// ---------------------------------------------------------------- degree
__global__ void k_degree(const int* __restrict__ col, float* __restrict__ deg) {
    int e = blockIdx.x * 256 + threadIdx.x;
    if (e < N_EDGES) atomicAdd(deg + col[e], 1.0f);
}

__global__ void k_dinv(float* __restrict__ deg) {
    int i = blockIdx.x * 256 + threadIdx.x;
    if (i < N_NODES) deg[i] = rsqrtf(deg[i]);   // deg >= 1 always (self-loop)
}

// ---------------------------------------------------------------- dual FP32 WMMA GEMM
// h   = xl @ gcn_w
// acc = xl @ rs_w + gcn_b + rs_b + dinv[i]^2 * h[i]   (self-loop folded in)
// TDM stages the 16x128 A tile and both 128x128 weight matrices into LDS
// (row stride 130 dwords via TDM padding), then the K-loop is pure LDS+WMMA.
__global__ void __launch_bounds__(256) k_dual_gemm(
    const float* __restrict__ xl, const float* __restrict__ gw,
    const float* __restrict__ rw, const float* __restrict__ gcn_b,
    const float* __restrict__ rs_b, const float* __restrict__ dinv,
    float* __restrict__ h, float* __restrict__ acc) {
    extern __shared__ float smem[];

    const int lane = threadIdx.x & 31;
    const int wave = threadIdx.x >> 5;
    const int m    = lane & 15;       // A: lane -> row M   | B/D: lane -> col N
    const int sel  = lane >> 4;       // K half select (A) / M half select (D)
    const int row0 = blockIdx.x * 16;
    const int n0   = wave * 16;

    if (threadIdx.x < 32) {           // wave 0 drives the Tensor Data Mover
        const unsigned int lds0 = (unsigned int)(uintptr_t)&smem[0];
        tdm_load_2d_f32(lds0 + SM_A * 4, xl + (size_t)row0 * D, D, 16, D, 16, D);
        tdm_load_2d_f32(lds0 + SM_G * 4, gw, D, D, D, D, D);
        tdm_load_2d_f32(lds0 + SM_R * 4, rw, D, D, D, D, D);
        __builtin_amdgcn_s_wait_tensorcnt(0);
    }
    __syncthreads();

    v8f ch = {0.f, 0.f, 0.f, 0.f, 0.f, 0.f, 0.f, 0.f};
    v8f cr = {0.f, 0.f, 0.f, 0.f, 0.f, 0.f, 0.f, 0.f};

    const float* sa = smem + SM_A + m * ROWSTRIDE;   // A row for this lane
    const float* sg = smem + SM_G + n0 + m;          // B column for this lane
    const float* sr = smem + SM_R + n0 + m;

#pragma unroll 4
    for (int k0 = 0; k0 < D; k0 += 4) {
        const int ka = k0 + 2 * sel;
        // A fragment 16x4 fp32: VGPR0 = A[m][ka], VGPR1 = A[m][ka+1]  (ISA 7.12.2)
        v2f a = *(const v2f*)(sa + ka);                 // 8B-aligned ds_load_b64
        // B fragment 4x16 fp32 (transpose-symmetric layout): lane holds column n0+m
        v2f bg, br;
        bg.x = sg[(size_t)ka * ROWSTRIDE];
        bg.y = sg[(size_t)(ka + 1) * ROWSTRIDE];
        br.x = sr[(size_t)ka * ROWSTRIDE];
        br.y = sr[(size_t)(ka + 1) * ROWSTRIDE];
        ch = __builtin_amdgcn_wmma_f32_16x16x4_f32(false, a, false, bg, (short)0, ch, false, false);
        cr = __builtin_amdgcn_wmma_f32_16x16x4_f32(false, a, false, br, (short)0, cr, false, false);
    }

    // D layout: VGPR r -> M = r + 8*sel, N = lane&15
    const float bsum = gcn_b[n0 + m] + rs_b[n0 + m];
#pragma unroll
    for (int r = 0; r < 8; ++r) {
        const int mrow = row0 + r + 8 * sel;
        const float di = dinv[mrow];
        const float hv = ch[r];
        h[(size_t)mrow * D + n0 + m]   = hv;
        acc[(size_t)mrow * D + n0 + m] = cr[r] + bsum + di * di * hv;
    }
}

// ---------------------------------------------------------------- edge scatter
// one wave32 per edge; lane handles 4 consecutive features
__global__ void __launch_bounds__(256) k_scatter(
    const int* __restrict__ rows, const int* __restrict__ cols,
    const float* __restrict__ dinv, const float* __restrict__ h,
    float* __restrict__ acc) {
    const int e    = blockIdx.x * 8 + (threadIdx.x >> 5);
    const int lane = threadIdx.x & 31;
    const int r = rows[e];
    const int c = cols[e];
    const float w = dinv[r] * dinv[c];
    const float4 v = ((const float4*)(h + (size_t)r * D))[lane];
    float* ap = acc + (size_t)c * D + lane * 4;
    atomicAdd(ap + 0, w * v.x);
    atomicAdd(ap + 1, w * v.y);
    atomicAdd(ap + 2, w * v.z);
    atomicAdd(ap + 3, w * v.w);
}

// ---------------------------------------------------------------- BatchNorm column reduction
__global__ void __launch_bounds__(256) k_bn_reduce(
    const float* __restrict__ acc, float* __restrict__ bnsum,
    float* __restrict__ bnsq) {
    const int d = threadIdx.x & 127;
    const int rowStart  = blockIdx.x * 2 + (threadIdx.x >> 7);
    const int rowStride = gridDim.x * 2;
    float s = 0.0f, s2 = 0.0f;
    for (int row = rowStart; row < N_NODES; row += rowStride) {
        const float v = acc[(size_t)row * D + d];
        s += v;
        s2 += v * v;
    }
    atomicAdd(bnsum + d, s);
    atomicAdd(bnsq + d, s2);
}

// ---------------------------------------------------------------- BN finalize + LeakyReLU
__global__ void __launch_bounds__(256) k_bn_act(
    float* __restrict__ out, const float* __restrict__ bnsum,
    const float* __restrict__ bnsq, const float* __restrict__ bn_g,
    const float* __restrict__ bn_b) {
    const int t = blockIdx.x * 256 + threadIdx.x;   // one float4 per thread
    const int d = (t * 4) & (D - 1);
    const float inv_n = 1.0f / (float)N_NODES;
    float4 v = ((const float4*)out)[t];
    float vv[4] = {v.x, v.y, v.z, v.w};
    float res[4];
#pragma unroll
    for (int j = 0; j < 4; ++j) {
        const float mu  = bnsum[d + j] * inv_n;
        const float var = bnsq[d + j] * inv_n - mu * mu;
        const float o = bn_g[d + j] * (vv[j] - mu) * rsqrtf(var + EPS_BN) + bn_b[d + j];
        res[j] = (o >= 0.0f) ? o : SLOPE * o;
    }
    ((float4*)out)[t] = make_float4(res[0], res[1], res[2], res[3]);
}

// ---------------------------------------------------------------- launcher
extern "C" void kernel_launch(void* const* d_in, const int* in_sizes, int n_in,
                              void* d_out, int out_size, void* d_ws, size_t ws_size,
                              hipStream_t stream) {
    (void)in_sizes; (void)n_in; (void)out_size; (void)ws_size;

    const float* x     = (const float*)d_in[0];
    const int*   edge  = (const int*)  d_in[1];   // [2, N_EDGES]: rows then cols
    const float* gcn_w = (const float*)d_in[2];
    const float* gcn_b = (const float*)d_in[3];
    const float* rs_w  = (const float*)d_in[4];
    const float* rs_b  = (const float*)d_in[5];
    const float* ln_g  = (const float*)d_in[6];
    const float* ln_b  = (const float*)d_in[7];
    const float* bn_g  = (const float*)d_in[8];
    const float* bn_b  = (const float*)d_in[9];
    float* out = (float*)d_out;

    // workspace layout (needs ~51.5 MB): xl | h | deg/dinv | bn sums
    char* ws = (char*)d_ws;
    const size_t FEAT = (size_t)N_NODES * D * sizeof(float);     // 25.6 MB
    float* xl    = (float*)(ws);
    float* h     = (float*)(ws + FEAT);
    float* deg   = (float*)(ws + 2 * FEAT);
    float* bnsum = (float*)(ws + 2 * FEAT + (((size_t)N_NODES * 4 + 255) & ~(size_t)255));
    float* bnsq  = bnsum + D;
    float* acc   = out;   // accumulate pre-BN result directly in d_out

    const int* erow = edge;
    const int* ecol = edge + N_EDGES;

    // opt-in to >64KB dynamic LDS (CDNA5 WGP has 320KB); host-side, capture-safe
    (void)hipFuncSetAttribute((const void*)k_dual_gemm,
                              hipFuncAttributeMaxDynamicSharedMemorySize,
                              SM_TOTAL_BYTES);

    k_init     <<<(N_NODES + 255) / 256, 256, 0, stream>>>(deg, bnsum, bnsq);
    k_layernorm<<<N_NODES / 8,           256, 0, stream>>>(x, ln_g, ln_b, xl);
    k_degree   <<<(N_EDGES + 255) / 256, 256, 0, stream>>>(ecol, deg);
    k_dinv     <<<(N_NODES + 255) / 256, 256, 0, stream>>>(deg);
    k_dual_gemm<<<N_NODES / 16, 256, SM_TOTAL_BYTES, stream>>>(xl, gcn_w, rs_w,
                                                               gcn_b, rs_b, deg, h, acc);
    k_scatter  <<<N_EDGES / 8,           256, 0, stream>>>(erow, ecol, deg, h, acc);
    k_bn_reduce<<<256,                   256, 0, stream>>>(acc, bnsum, bnsq);
    k_bn_act   <<<(N_NODES * (D / 4)) / 256, 256, 0, stream>>>(out, bnsum, bnsq, bn_g, bn_b);
}